// CTCLayer_46694884442631
// MI455X (gfx1250) — compile-verified
//
#include <hip/hip_runtime.h>
#include <stdint.h>

#ifndef __has_builtin
#define __has_builtin(x) 0
#endif

#define EPS_F     1e-7f
#define NEG_INF_F (-1e30f)
#define LN2_F     0.69314718055994530942f
#define CDIM      256
#define WPB       4    // waves (batches) per block
#define NBUF      16   // prefetch ring depth (rows in flight per wave)

typedef int v4i __attribute__((vector_size(16)));
typedef __attribute__((address_space(1))) void as1_void;
typedef __attribute__((address_space(3))) void as3_void;
typedef __attribute__((address_space(1))) v4i  as1_v4i;
typedef __attribute__((address_space(3))) v4i  as3_v4i;

// Async global -> LDS copy, 16 bytes per lane. IOFFSET applies to both the
// global and LDS addresses per the ISA, so one imm offset covers both sides.
#if __has_builtin(__builtin_amdgcn_global_load_async_to_lds_b128)
#define ASYNC_B128(gsrc, ldst, off)                                           \
  __builtin_amdgcn_global_load_async_to_lds_b128(                             \
      (as1_v4i*)(as1_void*)(gsrc), (as3_v4i*)(as3_void*)(ldst), (off), 0)
#else
#define ASYNC_B128(gsrc, ldst, off)                                           \
  asm volatile("global_load_async_to_lds_b128 %0, %1, off offset:" #off       \
               :: "v"((uint32_t)(uintptr_t)(ldst)), "v"(gsrc) : "memory")
#endif

#if __has_builtin(__builtin_amdgcn_s_wait_asynccnt)
#define WAIT_ASYNC(n)                                                         \
  do { __builtin_amdgcn_s_wait_asynccnt(n); asm volatile("" ::: "memory"); } while (0)
#else
#define WAIT_ASYNC(n) asm volatile("s_wait_asynccnt %0" :: "n"(n) : "memory")
#endif

// Single-instruction base-2 exp/log (v_exp_f32 / v_log_f32 are natively base-2).
__device__ __forceinline__ float fexp2(float x) {
#if __has_builtin(__builtin_amdgcn_exp2f)
  return __builtin_amdgcn_exp2f(x);
#else
  return exp2f(x);
#endif
}
__device__ __forceinline__ float flog2(float x) {
#if __has_builtin(__builtin_amdgcn_logf)
  return __builtin_amdgcn_logf(x);   // llvm.amdgcn.log == log2
#else
  return log2f(x);
#endif
}

// log-sum-exp in log2 domain (exact rescaling of the natural-log recurrence).
__device__ __forceinline__ float lse2(float a, float b) {
  float m = fmaxf(a, b);
  return m + flog2(fexp2(a - m) + fexp2(b - m));
}
__device__ __forceinline__ float lse3(float a, float b, float c) {
  float m = fmaxf(fmaxf(a, b), c);
  return m + flog2(fexp2(a - m) + fexp2(b - m) + fexp2(c - m));
}

// One wave32 per batch. Lane l owns states s = 4l..4l+3; lane 31 also owns
// s = 128 (always blank). Even states are blank (never skip); odd state
// 4l+1 uses label[2l], odd state 4l+3 uses label[2l+1].
__global__ __launch_bounds__(32 * WPB)
void ctc_wave_kernel(const int* __restrict__ y_true,       // B x L (L==64)
                     const float* __restrict__ y_pred,     // B x T x 256
                     const int* __restrict__ input_length, // B
                     const int* __restrict__ label_length, // B
                     float* __restrict__ out,              // B
                     int B, int T, int L) {
  __shared__ alignas(16) float rows[WPB][NBUF][CDIM];   // 16 KB per wave
  const int wave = threadIdx.x >> 5;
  const int lane = threadIdx.x & 31;
  const int b    = blockIdx.x * WPB + wave;
  if (b >= B) return;

  // Labels: lane l loads label[2l], label[2l+1] (coalesced).
  const int lab_lo   = y_true[(size_t)b * L + 2 * lane];
  const int lab_hi   = y_true[(size_t)b * L + 2 * lane + 1];
  const int lab_prev = __shfl_up(lab_hi, 1);                // label[2l-1]
  const bool skip1 = (lane > 0) && (lab_lo != lab_prev);    // s=4l+1 needs s>=2
  const bool skip3 = (lab_hi != lab_lo);                    // s=4l+3 (>=3 always)

  const int ilen = input_length[b];
  float* base = &rows[wave][0][0];

  // Deep prefetch prologue: issue rows 0..NBUF-1 (2 x b128 per lane per row).
  const float* gcur = y_pred + (size_t)b * T * CDIM + lane * 4;  // row 0, lane slice
  const int npre = (T < NBUF) ? T : NBUF;
  for (int r = 0; r < npre; ++r) {
    float* ls = base + r * CDIM + lane * 4;
    ASYNC_B128(gcur, ls, 0);
    ASYNC_B128(gcur, ls, 512);
    gcur += CDIM;                      // ends at row NBUF
  }

  float a0 = NEG_INF_F, a1 = NEG_INF_F, a2 = NEG_INF_F,
        a3 = NEG_INF_F, a4 = NEG_INF_F;

  for (int t = 0; t < T; ++t) {
    // Steady state: rows t+1..t+NBUF-1 (2*(NBUF-1) requests) may stay in
    // flight; completion is in-order so row t is then resident. Tail: drain.
    if (T - t > NBUF) { WAIT_ASYNC(2 * (NBUF - 1)); } else { WAIT_ASYNC(0); }

    const float* rb = base + (t & (NBUF - 1)) * CDIM;
    float pb = flog2(rb[CDIM - 1] + EPS_F);  // blank, LDS broadcast
    float p1 = flog2(rb[lab_lo] + EPS_F);
    float p3 = flog2(rb[lab_hi] + EPS_F);

    if (t == 0) {
      a0 = pb + ((lane == 0) ? 0.0f : NEG_INF_F);  // s = 4l (s<2 only lane 0)
      a1 = p1 + ((lane == 0) ? 0.0f : NEG_INF_F);  // s = 4l+1
      a2 = pb + NEG_INF_F;                         // s = 4l+2
      a3 = p3 + NEG_INF_F;                         // s = 4l+3
      a4 = pb + NEG_INF_F;                         // s = 128 (lane 31 only)
    } else {
      float sm1 = __shfl_up(a3, 1);  // alpha[4l-1] from lane l-1
      float sm2 = __shfl_up(a2, 1);  // alpha[4l-2] from lane l-1
      if (lane == 0) { sm1 = NEG_INF_F; sm2 = NEG_INF_F; }

      float n0 = pb + lse2(a0, sm1);
      float n1 = p1 + lse3(a1, a0, skip1 ? sm2 : NEG_INF_F);
      float n2 = pb + lse2(a2, a1);
      float n3 = p3 + lse3(a3, a2, skip3 ? a1 : NEG_INF_F);
      float n4 = pb + lse2(a4, a3);  // s=128, meaningful on lane 31 only

      if (t < ilen) { a0 = n0; a1 = n1; a2 = n2; a3 = n3; a4 = n4; }
    }

    // Refill the slot just consumed with row t+NBUF (issued after the LDS
    // gathers above were consumed, so the DMA cannot race the reads).
    if (t + NBUF < T) {
      float* ls = base + (t & (NBUF - 1)) * CDIM + lane * 4;
      ASYNC_B128(gcur, ls, 0);
      ASYNC_B128(gcur, ls, 512);
      gcur += CDIM;
    }
  }

  // Gather final alphas into a dead row buffer (LDS ops are in-order per wave).
  float* fin = base;
  fin[4 * lane + 0] = a0;
  fin[4 * lane + 1] = a1;
  fin[4 * lane + 2] = a2;
  fin[4 * lane + 3] = a3;
  if (lane == 31) fin[128] = a4;
  asm volatile("" ::: "memory");
  if (lane == 0) {
    const int end = 2 * label_length[b];           // in [64, 128]
    float al = fin[end];
    float ap = fin[end - 1];
    out[b] = -(lse2(al, ap) * LN2_F);              // back to natural log
  }
}

extern "C" void kernel_launch(void* const* d_in, const int* in_sizes, int n_in,
                              void* d_out, int out_size, void* d_ws, size_t ws_size,
                              hipStream_t stream) {
  const int*   y_true = (const int*)d_in[0];
  const float* y_pred = (const float*)d_in[1];
  const int*   ilen   = (const int*)d_in[2];
  const int*   llen   = (const int*)d_in[3];
  float*       out    = (float*)d_out;

  const int B = in_sizes[2];                 // 1024
  const int L = in_sizes[0] / B;             // 64
  const int T = in_sizes[1] / (B * CDIM);    // 512

  const int blocks = (B + WPB - 1) / WPB;    // 256 blocks x 4 waves
  ctc_wave_kernel<<<blocks, 32 * WPB, 0, stream>>>(y_true, y_pred, ilen, llen,
                                                   out, B, T, L);
}